// DeeperGCNBase_14697378087225
// MI455X (gfx1250) — compile-verified
//
#include <hip/hip_runtime.h>
#include <hip/hip_bf16.h>
#include <math.h>

// ---------------------------------------------------------------------------
// DeeperGCN (GENConv softmax-aggr, 2 layers) for MI455X / gfx1250, fp32.
// GEMMs use V_WMMA_F32_16X16X4_F32 (true fp32 matrix core).
// Edge aggregation: 2 passes, atomics into L2-resident [N,128] buffers.
// ---------------------------------------------------------------------------

#define DD   128            // hidden dim D
#define DD2  256            // MLP dim D2
#define MSG_EPS 1e-7f

typedef __attribute__((ext_vector_type(2))) float v2f;
typedef __attribute__((ext_vector_type(8))) float v8f;

// ---------------- generic fill ----------------
__global__ void k_fill(float* __restrict__ p, float v, long n) {
    long i = (long)blockIdx.x * blockDim.x + threadIdx.x;
    if (i < n) p[i] = v;
}

// ---------------- WMMA fp32 GEMM:  C[M,Nc] = A[M,K] @ W[K,Nc] + bias ----------------
// One wave per 16x16 output tile; 4 waves/block share the same 16 A-rows
// (A re-read hits L1/L2). M, K, Nc all multiples of 16; M multiple of 16.
//
// A-frag (16x4 f32, ISA 7.12.2): lane<16 -> row=lane,  a={A[r][k0],A[r][k0+1]}
//                                lane>=16 -> row=l-16, a={A[r][k0+2],A[r][k0+3]}
// B-frag (4x16): lane<16 -> b={W[k0][n],W[k0+1][n]}, lane>=16 -> K+2/K+3.
// C/D   : vgpr v, lanes0-15: (M=v, N=lane); lanes16-31: (M=v+8, N=lane-16).
__global__ void k_gemm_wmma_f32(const float* __restrict__ A,
                                const float* __restrict__ W,
                                const float* __restrict__ bias,
                                float* __restrict__ C,
                                int M, int K, int Nc) {
    const int lane = threadIdx.x & 31;
    const int wave = threadIdx.x >> 5;
    const int row0 = blockIdx.y << 4;
    const int col0 = (blockIdx.x << 6) + (wave << 4);
    if (col0 >= Nc) return;                       // whole-wave uniform exit
    const int half = lane >> 4;                   // 0: lanes 0-15, 1: lanes 16-31
    const int lr   = lane & 15;

    const float* __restrict__ Arow = A + (size_t)(row0 + lr) * K;
    v8f acc = {};
    for (int k0 = 0; k0 < K; k0 += 4) {
        const int ka = k0 + (half << 1);
        v2f a, b;
        a.x = Arow[ka];
        a.y = Arow[ka + 1];
        b.x = W[(size_t)ka       * Nc + col0 + lr];
        b.y = W[(size_t)(ka + 1) * Nc + col0 + lr];
        acc = __builtin_amdgcn_wmma_f32_16x16x4_f32(
                  /*neg_a=*/false, a, /*neg_b=*/false, b,
                  /*c_mod=*/(short)0, acc, /*reuse_a=*/false, /*reuse_b=*/false);
    }
    const float bc = bias[col0 + lr];
    #pragma unroll
    for (int v = 0; v < 8; ++v) {
        const int m = row0 + v + (half << 3);
        C[(size_t)m * Nc + col0 + lr] = acc[v] + bc;
    }
}

// ---------------- float atomic max (sign-split trick, any sign) ----------------
__device__ __forceinline__ void atomicMaxF(float* addr, float val) {
    if (val >= 0.0f) atomicMax((int*)addr, __float_as_int(val));
    else             atomicMin((unsigned int*)addr, __float_as_uint(val));
}

// ---------------- edge pass 1: per-dst segment max of logits ----------------
// One wave per edge; lane handles 4 features (float4).
__global__ void k_edge_max(const float* __restrict__ h,
                           const int* __restrict__ src,
                           const int* __restrict__ dst,
                           const float* __restrict__ tptr,
                           float* __restrict__ mx, int E) {
    const int gid  = blockIdx.x * blockDim.x + threadIdx.x;
    const int e    = gid >> 5;
    const int lane = gid & 31;
    if (e >= E) return;
    const float t = tptr[0];
    const int s = src[e], d = dst[e];
    const float4 xv = *reinterpret_cast<const float4*>(h + (size_t)s * DD + lane * 4);
    float* mp = mx + (size_t)d * DD + lane * 4;
    atomicMaxF(mp + 0, (fmaxf(xv.x, 0.f) + MSG_EPS) * t);
    atomicMaxF(mp + 1, (fmaxf(xv.y, 0.f) + MSG_EPS) * t);
    atomicMaxF(mp + 2, (fmaxf(xv.z, 0.f) + MSG_EPS) * t);
    atomicMaxF(mp + 3, (fmaxf(xv.w, 0.f) + MSG_EPS) * t);
}

// ---------------- edge pass 2: fused exp-sum + weighted-sum ----------------
__global__ void k_edge_accum(const float* __restrict__ h,
                             const int* __restrict__ src,
                             const int* __restrict__ dst,
                             const float* __restrict__ tptr,
                             const float* __restrict__ mx,
                             float* __restrict__ ssum,
                             float* __restrict__ wsum, int E) {
    const int gid  = blockIdx.x * blockDim.x + threadIdx.x;
    const int e    = gid >> 5;
    const int lane = gid & 31;
    if (e >= E) return;
    const float t = tptr[0];
    const int s = src[e], d = dst[e];
    const size_t off = (size_t)d * DD + lane * 4;
    const float4 xv = *reinterpret_cast<const float4*>(h + (size_t)s * DD + lane * 4);
    const float4 mv = *reinterpret_cast<const float4*>(mx + off);
    float msg[4] = { fmaxf(xv.x, 0.f) + MSG_EPS, fmaxf(xv.y, 0.f) + MSG_EPS,
                     fmaxf(xv.z, 0.f) + MSG_EPS, fmaxf(xv.w, 0.f) + MSG_EPS };
    float m4[4]  = { mv.x, mv.y, mv.z, mv.w };
    #pragma unroll
    for (int i = 0; i < 4; ++i) {
        const float ev = __expf(msg[i] * t - m4[i]);
        atomicAdd(ssum + off + i, ev);
        atomicAdd(wsum + off + i, ev * msg[i]);
    }
}

// ---------------- agg finalize + root add:  m0 = wsum/(ssum+1e-16) + h ----------------
__global__ void k_combine(const float* __restrict__ wsum,
                          const float* __restrict__ ssum,
                          const float* __restrict__ h,
                          float* __restrict__ m0, long n) {
    long i = (long)blockIdx.x * blockDim.x + threadIdx.x;
    if (i < n) m0[i] = wsum[i] / (ssum[i] + 1e-16f) + h[i];
}

// ---------------- batchnorm column reduction (blockDim.x == DD2) ----------------
__global__ void k_col_sums(const float* __restrict__ H, float* __restrict__ s1,
                           float* __restrict__ s2, int M, int rowsPerBlock) {
    const int c  = threadIdx.x;                    // column, coalesced across threads
    const int r0 = blockIdx.x * rowsPerBlock;
    const int r1 = min(r0 + rowsPerBlock, M);
    float a = 0.f, b = 0.f;
    for (int r = r0; r < r1; ++r) {
        const float v = H[(size_t)r * DD2 + c];
        a += v; b += v * v;
    }
    atomicAdd(s1 + c, a);
    atomicAdd(s2 + c, b);
}

// ---------------- batchnorm apply + relu (in place) ----------------
__global__ void k_bn_relu(float* __restrict__ H, const float* __restrict__ s1,
                          const float* __restrict__ s2, const float* __restrict__ g,
                          const float* __restrict__ b, int M) {
    long i = (long)blockIdx.x * blockDim.x + threadIdx.x;
    if (i >= (long)M * DD2) return;
    const int c = (int)(i & (DD2 - 1));
    const float invM = 1.0f / (float)M;
    const float mu  = s1[c] * invM;
    const float var = s2[c] * invM - mu * mu;
    const float r   = rsqrtf(var + 1e-5f);
    const float v   = (H[i] - mu) * r * g[c] + b[c];
    H[i] = v > 0.f ? v : 0.f;
}

// ---------------- layernorm + relu + residual (one wave per row, wave32) ----------------
__global__ void k_ln_relu_res(const float* __restrict__ O, const float* __restrict__ g,
                              const float* __restrict__ b, float* __restrict__ h, int M) {
    const int gid  = blockIdx.x * blockDim.x + threadIdx.x;
    const int row  = gid >> 5;
    const int lane = gid & 31;
    if (row >= M) return;
    const size_t off = (size_t)row * DD + lane * 4;
    const float4 v = *reinterpret_cast<const float4*>(O + off);
    float s = v.x + v.y + v.z + v.w;
    float q = v.x * v.x + v.y * v.y + v.z * v.z + v.w * v.w;
    #pragma unroll
    for (int m = 16; m; m >>= 1) {      // wave32 butterfly reduction
        s += __shfl_xor(s, m, 32);
        q += __shfl_xor(q, m, 32);
    }
    const float mu  = s * (1.0f / DD);
    const float var = q * (1.0f / DD) - mu * mu;
    const float r   = rsqrtf(var + 1e-5f);
    const float4 gg = *reinterpret_cast<const float4*>(g + lane * 4);
    const float4 bb = *reinterpret_cast<const float4*>(b + lane * 4);
    float4 hv = *reinterpret_cast<const float4*>(h + off);
    hv.x += fmaxf((v.x - mu) * r * gg.x + bb.x, 0.f);
    hv.y += fmaxf((v.y - mu) * r * gg.y + bb.y, 0.f);
    hv.z += fmaxf((v.z - mu) * r * gg.z + bb.z, 0.f);
    hv.w += fmaxf((v.w - mu) * r * gg.w + bb.w, 0.f);
    *reinterpret_cast<float4*>(h + off) = hv;
}

// ---------------------------------------------------------------------------
static inline void launch_gemm(const float* A, const float* W, const float* bias,
                               float* C, int M, int K, int Nc, hipStream_t st) {
    dim3 grid((Nc + 63) / 64, M / 16);
    k_gemm_wmma_f32<<<grid, 128, 0, st>>>(A, W, bias, C, M, K, Nc);
}

extern "C" void kernel_launch(void* const* d_in, const int* in_sizes, int n_in,
                              void* d_out, int out_size, void* d_ws, size_t ws_size,
                              hipStream_t stream) {
    const float* x     = (const float*)d_in[0];
    const int*   eidx  = (const int*)  d_in[1];
    const float* in_w  = (const float*)d_in[2];
    const float* in_b  = (const float*)d_in[3];
    const float* t     = (const float*)d_in[4];
    const float* w1    = (const float*)d_in[5];
    const float* b1    = (const float*)d_in[6];
    const float* bn_g  = (const float*)d_in[7];
    const float* bn_b  = (const float*)d_in[8];
    const float* w2    = (const float*)d_in[9];
    const float* b2    = (const float*)d_in[10];
    const float* ln_g  = (const float*)d_in[11];
    const float* ln_b  = (const float*)d_in[12];
    const float* out_w = (const float*)d_in[13];
    const float* out_b = (const float*)d_in[14];

    const int N = in_sizes[0] / DD;        // 50000 (multiple of 16)
    const int E = in_sizes[1] / 2;         // 640000
    const int L = in_sizes[4];             // 2
    const int C_OUT = in_sizes[14];        // 64
    const int* src = eidx;
    const int* dst = eidx + E;

    // ---- workspace carve-up (all fp32) ----
    const size_t nD = (size_t)N * DD;
    float* h    = (float*)d_ws;            // [N,128] running node features
    float* mx   = h    + nD;               // [N,128] segment max; reused as m0
    float* ssum = mx   + nD;               // [N,128] exp-sum
    float* wsum = ssum + nD;               // [N,128] weighted sum; reused as h2
    float* h1   = wsum + nD;               // [N,256] MLP hidden
    float* bns1 = h1 + (size_t)N * DD2;    // [256]
    float* bns2 = bns1 + DD2;              // [256]

    const long nElemD  = (long)N * DD;
    const long nElemD2 = (long)N * DD2;
    const int TPB = 256;
    const dim3 gD((unsigned)((nElemD + TPB - 1) / TPB));
    const dim3 gD2((unsigned)((nElemD2 + TPB - 1) / TPB));
    const dim3 gEdge((unsigned)(((long)E * 32 + TPB - 1) / TPB));
    const dim3 gRow((unsigned)(((long)N * 32 + TPB - 1) / TPB));

    // ---- input projection: h = x @ in_w + in_b ----
    launch_gemm(x, in_w, in_b, h, N, DD, DD, stream);

    for (int l = 0; l < L; ++l) {
        // init aggregation buffers
        k_fill<<<gD, TPB, 0, stream>>>(mx,  -3.0e38f, nElemD);
        k_fill<<<gD, TPB, 0, stream>>>(ssum, 0.0f,    nElemD);
        k_fill<<<gD, TPB, 0, stream>>>(wsum, 0.0f,    nElemD);

        // softmax aggregation (2 edge passes, atomics stay in the 192MB L2)
        k_edge_max  <<<gEdge, TPB, 0, stream>>>(h, src, dst, t + l, mx, E);
        k_edge_accum<<<gEdge, TPB, 0, stream>>>(h, src, dst, t + l, mx, ssum, wsum, E);

        // m0 = agg + h   (written into mx buffer, now free)
        k_combine<<<gD, TPB, 0, stream>>>(wsum, ssum, h, mx, nElemD);

        // MLP layer 1: h1 = m0 @ w1[l] + b1[l]
        launch_gemm(mx, w1 + (size_t)l * DD * DD2, b1 + (size_t)l * DD2,
                    h1, N, DD, DD2, stream);

        // batchnorm (batch stats) + relu, in place
        k_fill<<<dim3(2), TPB, 0, stream>>>(bns1, 0.0f, 2 * DD2);
        k_col_sums<<<dim3((N + 255) / 256), DD2, 0, stream>>>(h1, bns1, bns2, N, 256);
        k_bn_relu<<<gD2, TPB, 0, stream>>>(h1, bns1, bns2,
                                           bn_g + (size_t)l * DD2, bn_b + (size_t)l * DD2, N);

        // MLP layer 2: h2 = h1 @ w2[l] + b2[l]   (h2 stored in wsum buffer)
        launch_gemm(h1, w2 + (size_t)l * DD2 * DD, b2 + (size_t)l * DD,
                    wsum, N, DD2, DD, stream);

        // layernorm + relu + residual into h
        k_ln_relu_res<<<gRow, TPB, 0, stream>>>(wsum, ln_g + (size_t)l * DD,
                                                ln_b + (size_t)l * DD, h, N);
    }

    // ---- output projection: out = h @ out_w + out_b ----
    launch_gemm(h, out_w, out_b, (float*)d_out, N, DD, C_OUT, stream);
}